// Customized_45990509805590
// MI455X (gfx1250) — compile-verified
//
#include <hip/hip_runtime.h>
#include <stdint.h>

// ---------------------------------------------------------------------------
// 101 x (per-pixel scalar affine + ReLU) over 32x1x1024x1024 fp32.
// Strategy:
//   1) collapse_chain_kernel: fold the 101 layers into f(x)=max(C, A*x+D)
//      (valid for all w_i >= 0, which holds for w = 1 + 0.02*N(0,1));
//      writes {A, D, C, negflag} to d_ws.
//   2) stream_affine_kernel: memory-bound streaming pass. One wave32 per
//      workgroup issues TENSOR_LOAD_TO_LDS (TDM) for 8KB tiles into
//      ping-pong LDS buffers, pipelined with s_wait_tensorcnt, then does
//      128-bit LDS reads -> 1 FMA + 1 max -> 128-bit global stores.
//      If negflag is set (some w<0), a uniform branch runs the literal
//      101-step chain instead (w,b cached in LDS).
// Roofline: 268 MB @ 23.3 TB/s ~ 11.5 us; collapsed math (2 ops/elt) keeps
// this purely HBM-bound.
// ---------------------------------------------------------------------------

#define TILE     2048      // fp32 elements per TDM tile (8 KB)
#define MAXLAY   128
#define NBLOCKS  4096

typedef unsigned int u32x4 __attribute__((ext_vector_type(4)));
typedef int          i32x4 __attribute__((ext_vector_type(4)));
typedef int          i32x8 __attribute__((ext_vector_type(8)));

#ifndef __has_builtin
#define __has_builtin(x) 0
#endif

#if defined(__AMDGCN__) && defined(__gfx1250__) && \
    __has_builtin(__builtin_amdgcn_tensor_load_to_lds) && \
    __has_builtin(__builtin_amdgcn_s_wait_tensorcnt)
#define USE_TDM 1
#else
#define USE_TDM 0
#endif

// ---------------------------------------------------------------------------
// Kernel 1: fold the affine chain. f_i = g_i o f_{i-1}, maintained as
// f(x) = max(C, A*x + D):  A' = w*A, D' = w*D + b, C' = max(0, w*C + b).
// ---------------------------------------------------------------------------
__global__ void collapse_chain_kernel(const float* __restrict__ w,
                                      const float* __restrict__ b,
                                      int nlayers,
                                      float* __restrict__ acd) {
  if (threadIdx.x == 0 && blockIdx.x == 0) {
    float A = 1.0f;
    float D = 0.0f;
    float C = -3.402823466e38f;  // -FLT_MAX: identity has no lower clamp
    int neg = 0;
    for (int i = 0; i < nlayers; ++i) {
      const float wi = w[i];
      const float bi = b[i];
      neg |= (wi < 0.0f) ? 1 : 0;
      C = fmaxf(0.0f, C * wi + bi);
      D = D * wi + bi;
      A = A * wi;
    }
    acd[0] = A;
    acd[1] = D;
    acd[2] = C;
    acd[3] = neg ? 1.0f : 0.0f;
  }
}

// ---------------------------------------------------------------------------
// TDM tile load: 2D tile (tile_dim0 = TILE, tile_dim1 = 1) of 4-byte elements
// from global `gptr` into LDS byte offset `lds_byte_off`.
// D# packing per CDNA5 ISA ch.8 (async/tensor):
//   group0: [1:0]=count=1, [63:32]=lds_addr, [120:64]=global_addr, [127:126]=2
//   group1: [17:16]=data_size(2 -> 4B), [79:48]=tensor_dim0, [111:80]=tensor_dim1,
//           [127:112]=tile_dim0, [143:128]=tile_dim1, [207:160]=dim0_stride
//   groups 2/3 + trailing group: zero (<=2D tensor, no gather/iterate)
// ---------------------------------------------------------------------------
#if USE_TDM
__device__ __forceinline__ void tdm_load_tile(const float* gptr,
                                              uint32_t lds_byte_off) {
  const uint64_t ga = (uint64_t)(uintptr_t)gptr;

  u32x4 g0;
  g0.x = 1u;                                             // count=1, is_restore=0
  g0.y = lds_byte_off;                                   // lds_addr (bytes)
  g0.z = (uint32_t)(ga & 0xffffffffull);                 // global_addr[31:0]
  g0.w = (uint32_t)((ga >> 32) & 0x01ffffffull)          // global_addr[56:32]
         | (2u << 30);                                   // type=2 ("image")

  i32x8 g1;
  g1[0] = (int)(2u << 16);                    // data_size=2 (4 bytes), mask=0
  g1[1] = (int)(((unsigned)TILE & 0xffffu) << 16);        // tensor_dim0[15:0]
  g1[2] = (int)((((unsigned)TILE >> 16) & 0xffffu)        // tensor_dim0[31:16]
                | (1u << 16));                            // tensor_dim1[15:0]=1
  g1[3] = (int)(((unsigned)TILE & 0xffffu) << 16);        // tile_dim0=TILE
  g1[4] = 1;                                              // tile_dim1=1, tile_dim2=0
  g1[5] = TILE;                                           // dim0_stride[31:0]
  g1[6] = 0;                                              // dim0_stride[47:32]
  g1[7] = 0;

  i32x4 gz4 = {0, 0, 0, 0};
  i32x8 gz8 = {0, 0, 0, 0, 0, 0, 0, 0};
  // clang-23 / therock-10.0 form: 6 args (g0, g1, g2, g3, extra, cpol)
  __builtin_amdgcn_tensor_load_to_lds(g0, g1, gz4, gz4, gz8, 0);
}
#endif

// ---------------------------------------------------------------------------
// Kernel 2: streaming pass. One wave32 per block; grid-stride over tiles with
// 2-deep TDM pipeline (ping-pong LDS).
// ---------------------------------------------------------------------------
__global__ __launch_bounds__(32)
void stream_affine_kernel(const float* __restrict__ x,
                          float* __restrict__ y,
                          const float* __restrict__ w,
                          const float* __restrict__ b,
                          const float* __restrict__ acd,
                          int n, int nlayers) {
  __shared__ __align__(16) float sbuf[2][TILE];
  __shared__ float sw[MAXLAY];
  __shared__ float sb[MAXLAY];

  const float A = acd[0];
  const float D = acd[1];
  const float C = acd[2];
  const bool slow = (acd[3] != 0.0f);   // uniform branch; ~never taken

  const int lane   = threadIdx.x;
  const int stride = gridDim.x;

  if (slow) {
    for (int i = lane; i < nlayers; i += 32) { sw[i] = w[i]; sb[i] = b[i]; }
    __syncthreads();
  }

  const int ntiles = n / TILE;
  constexpr int V4 = TILE / (4 * 32);   // float4 iterations per lane per tile

#if USE_TDM
  if ((int)blockIdx.x < ntiles)
    tdm_load_tile(x + (size_t)blockIdx.x * TILE,
                  (uint32_t)(uintptr_t)&sbuf[0][0]);

  int buf = 0;
  for (int t = (int)blockIdx.x; t < ntiles; t += stride) {
    const int tn = t + stride;
    if (tn < ntiles) {
      tdm_load_tile(x + (size_t)tn * TILE,
                    (uint32_t)(uintptr_t)&sbuf[buf ^ 1][0]);
      __builtin_amdgcn_s_wait_tensorcnt(1);   // tile t complete (in-order)
    } else {
      __builtin_amdgcn_s_wait_tensorcnt(0);
    }
    asm volatile("" ::: "memory");            // no LDS-read hoisting past wait

    const float4* __restrict__ src = (const float4*)&sbuf[buf][0];
    float4* __restrict__ dst = (float4*)(y + (size_t)t * TILE);

    if (!slow) {
#pragma unroll
      for (int k = 0; k < V4; ++k) {
        float4 v = src[k * 32 + lane];
        v.x = fmaxf(C, A * v.x + D);
        v.y = fmaxf(C, A * v.y + D);
        v.z = fmaxf(C, A * v.z + D);
        v.w = fmaxf(C, A * v.w + D);
        dst[k * 32 + lane] = v;
      }
    } else {
      for (int k = 0; k < V4; ++k) {
        float4 v = src[k * 32 + lane];
        for (int i = 0; i < nlayers; ++i) {
          const float wi = sw[i], bi = sb[i];
          v.x = fmaxf(v.x * wi + bi, 0.0f);
          v.y = fmaxf(v.y * wi + bi, 0.0f);
          v.z = fmaxf(v.z * wi + bi, 0.0f);
          v.w = fmaxf(v.w * wi + bi, 0.0f);
        }
        dst[k * 32 + lane] = v;
      }
    }
    buf ^= 1;
  }
#else
  // Fallback (no TDM builtin): direct 128-bit global streaming.
  const float4* __restrict__ xv = (const float4*)x;
  float4* __restrict__ yv = (float4*)y;
  const int n4 = (ntiles * TILE) / 4;
  for (int i = (int)blockIdx.x * 32 + lane; i < n4; i += stride * 32) {
    float4 v = xv[i];
    if (!slow) {
      v.x = fmaxf(C, A * v.x + D);
      v.y = fmaxf(C, A * v.y + D);
      v.z = fmaxf(C, A * v.z + D);
      v.w = fmaxf(C, A * v.w + D);
    } else {
      for (int j = 0; j < nlayers; ++j) {
        const float wi = sw[j], bi = sb[j];
        v.x = fmaxf(v.x * wi + bi, 0.0f);
        v.y = fmaxf(v.y * wi + bi, 0.0f);
        v.z = fmaxf(v.z * wi + bi, 0.0f);
        v.w = fmaxf(v.w * wi + bi, 0.0f);
      }
    }
    yv[i] = v;
  }
#endif

  // Tail elements (n not divisible by TILE; not hit for 32*1024*1024).
  for (int i = ntiles * TILE + (int)blockIdx.x * 32 + lane; i < n;
       i += stride * 32) {
    float v = x[i];
    if (!slow) {
      v = fmaxf(C, A * v + D);
    } else {
      for (int j = 0; j < nlayers; ++j)
        v = fmaxf(v * sw[j] + sb[j], 0.0f);
    }
    y[i] = v;
  }
}

// ---------------------------------------------------------------------------
extern "C" void kernel_launch(void* const* d_in, const int* in_sizes, int n_in,
                              void* d_out, int out_size, void* d_ws,
                              size_t ws_size, hipStream_t stream) {
  const float* x = (const float*)d_in[0];
  const float* w = (const float*)d_in[1];
  const float* b = (const float*)d_in[2];
  float* y   = (float*)d_out;
  float* acd = (float*)d_ws;   // {A, D, C, negflag}

  const int n = in_sizes[0];
  int nlayers = in_sizes[1];
  if (nlayers > MAXLAY) nlayers = MAXLAY;

  collapse_chain_kernel<<<1, 32, 0, stream>>>(w, b, nlayers, acd);

  int blocks = NBLOCKS;
  const int ntiles = n / TILE;
  if (blocks > ntiles && ntiles > 0) blocks = ntiles;
  if (blocks < 1) blocks = 1;
  stream_affine_kernel<<<blocks, 32, 0, stream>>>(x, y, w, b, acd, n, nlayers);
}